// GraphFeatureFusion_48498770707157
// MI455X (gfx1250) — compile-verified
//
#include <hip/hip_runtime.h>
#include <hip/hip_bf16.h>

typedef float v2f __attribute__((ext_vector_type(2)));
typedef float v8f __attribute__((ext_vector_type(8)));

// ---------- helpers ----------
__device__ __forceinline__ unsigned fmap(float f) {
    unsigned u = __float_as_uint(f);
    return (u & 0x80000000u) ? ~u : (u | 0x80000000u);
}
__device__ __forceinline__ float funmap(unsigned u) {
    return (u & 0x80000000u) ? __uint_as_float(u & 0x7FFFFFFFu) : __uint_as_float(~u);
}

// ---------- init: deg accumulator = 1 (self loop), denom=0, gmax=min, out=0 ----------
__global__ void init_kernel(float* deg, float* denom, unsigned* gmaxu, float* out,
                            int N, int B, int outN) {
    int i = blockIdx.x * blockDim.x + threadIdx.x;
    if (i < N) deg[i] = 1.0f;
    if (i < B) { denom[i] = 0.0f; gmaxu[i] = 0u; }
    if (i < outN) out[i] = 0.0f;
}

__global__ void deg_kernel(const int* __restrict__ dst, float* __restrict__ deg, int E) {
    int e = blockIdx.x * blockDim.x + threadIdx.x;
    if (e < E) atomicAdd(&deg[dst[e]], 1.0f);
}

__global__ void rsqrt_kernel(float* d, int N) {
    int i = blockIdx.x * blockDim.x + threadIdx.x;
    if (i < N) {
        float v = d[i];
        d[i] = (v > 0.0f) ? rsqrtf(v) : 0.0f;
    }
}

// ---------- fp32 WMMA GEMM with LDS-staged, double-buffered B tiles ----------
// C[M,Nout] = A[M,K] * W[K,Nout]
// block = 256 threads (8 waves). Workgroup tile: 128 rows x 64 cols.
// Each wave: 16 rows x 64 cols (4 n-tiles of 16). K consumed in chunks of KC=32.
//
// LDS holds W in *fragment order*: element W[k][col] lives at
//   bs[(k>>1)*PRS + col*2 + (k&1)]
// so each B fragment (two consecutive k rows, one col) is one 8-byte
// ds_load_b64 directly into the even VGPR pair the WMMA consumes.
// PRS=160 floats => pair-rows 1 apart differ by 32 banks => lanes 0-15
// (pair-row p) and lanes 16-31 (pair-row p+1) cover disjoint bank halves:
// conflict-free. Double buffered so global loads overlap compute.
#define GEMM_KC  32
#define GEMM_PRS 160   // floats per pair-row (80 pairs); 160 mod 64 == 32

__global__ __launch_bounds__(256) void gemm_wmma_f32(const float* __restrict__ A,
                                                     const float* __restrict__ W,
                                                     float* __restrict__ C,
                                                     int M, int K, int Nout) {
    __shared__ float bs[2][(GEMM_KC / 2) * GEMM_PRS];

    const int tid   = threadIdx.x;
    const int lane  = tid & 31;
    const int wave  = tid >> 5;
    const int l15   = lane & 15;
    const int lhalf = lane >> 4;             // 0 or 1
    const int mBase = blockIdx.x * 128 + wave * 16;
    const int nBase = blockIdx.y * 64;

    // cooperative W-stager mapping: thread -> pair-row rp (k rows 2rp,2rp+1), float4 col lc
    const int rp = tid >> 4;                 // 0..15
    const int lc = (tid & 15) << 2;          // 0,4,...,60

    v8f acc0 = {}, acc1 = {}, acc2 = {}, acc3 = {};

    int arow = mBase + l15;
    if (arow >= M) arow = M - 1;             // clamp (stores are guarded)
    const float* aptr = A + (size_t)arow * K + (lhalf << 1);

    const int nchunks = K / GEMM_KC;

    // prologue: fetch chunk 0 of W into regs (rows 2rp and 2rp+1)
    const float* wsrc = W + (size_t)(2 * rp) * Nout + nBase + lc;
    float4 w0 = *(const float4*)(wsrc);
    float4 w1 = *(const float4*)(wsrc + Nout);

    for (int c = 0; c < nchunks; ++c) {
        const int bsel = c & 1;
        // stage current chunk into LDS, k-pair interleaved (contiguous v2f pairs)
        {
            v2f* dp = (v2f*)&bs[bsel][rp * GEMM_PRS] + lc;
            v2f p0 = { w0.x, w1.x };
            v2f p1 = { w0.y, w1.y };
            v2f p2 = { w0.z, w1.z };
            v2f p3 = { w0.w, w1.w };
            dp[0] = p0; dp[1] = p1; dp[2] = p2; dp[3] = p3;
        }
        __syncthreads();

        // prefetch next chunk into regs (overlaps compute below)
        if (c + 1 < nchunks) {
            const float* wn = W + (size_t)((c + 1) * GEMM_KC + 2 * rp) * Nout + nBase + lc;
            w0 = *(const float4*)(wn);
            w1 = *(const float4*)(wn + Nout);
        }

        // compute 8 k-steps (KC=32) from LDS
#pragma unroll
        for (int kk = 0; kk < GEMM_KC; kk += 4) {
            v2f a = *(const v2f*)aptr;       // A[row][k..k+1] (per half-wave k pair)
            aptr += 4;
            // B fragment: pair-row (kk/2 + lhalf), one ds_load_b64 per n-tile
            const v2f* bp = (const v2f*)&bs[bsel][((kk >> 1) + lhalf) * GEMM_PRS] + l15;
            v2f b0 = bp[0];
            v2f b1 = bp[16];
            v2f b2 = bp[32];
            v2f b3 = bp[48];
            acc0 = __builtin_amdgcn_wmma_f32_16x16x4_f32(false, a, false, b0, (short)0, acc0, false, false);
            acc1 = __builtin_amdgcn_wmma_f32_16x16x4_f32(false, a, false, b1, (short)0, acc1, false, false);
            acc2 = __builtin_amdgcn_wmma_f32_16x16x4_f32(false, a, false, b2, (short)0, acc2, false, false);
            acc3 = __builtin_amdgcn_wmma_f32_16x16x4_f32(false, a, false, b3, (short)0, acc3, false, false);
        }
        __syncthreads();
    }

    // C/D layout: vgpr r -> row mBase + 8*lhalf + r, col nBase + l15 (+16 per tile)
    const int rowb = mBase + (lhalf << 3);
#pragma unroll
    for (int r = 0; r < 8; ++r) {
        int row = rowb + r;
        if (row < M) {
            float* cp = C + (size_t)row * Nout + nBase + l15;
            cp[0]  = acc0[r];
            cp[16] = acc1[r];
            cp[32] = acc2[r];
            cp[48] = acc3[r];
        }
    }
}

// ---------- self-loop init: agg[n,:] = xw[n,:] * dinv[n]^2 (over N*64 float4) ----------
__global__ void self_init_kernel(const float* __restrict__ xw, const float* __restrict__ dinv,
                                 float* __restrict__ agg, int N) {
    int i = blockIdx.x * blockDim.x + threadIdx.x;
    int total = N * 64;
    if (i < total) {
        int n = i >> 6;
        float c = dinv[n] * dinv[n];
        float4 v = ((const float4*)xw)[i];
        float4 o = { v.x * c, v.y * c, v.z * c, v.w * c };
        ((float4*)agg)[i] = o;
    }
}

// ---------- edge scatter-add: agg[d,:] += xw[s,:] * dinv[s]*dinv[d] ----------
__global__ __launch_bounds__(256) void edge_agg_kernel(const int* __restrict__ src,
                                                       const int* __restrict__ dst,
                                                       const float* __restrict__ dinv,
                                                       const float* __restrict__ xw,
                                                       float* __restrict__ agg, int E) {
    const int fq = (threadIdx.x & 63) << 2;  // feature offset 0..252
    const int eo = threadIdx.x >> 6;         // 0..3
    long base = (long)blockIdx.x * 16;
#pragma unroll
    for (int i = 0; i < 4; ++i) {
        long e = base + i * 4 + eo;
        if (e < E) {
            int s = src[e], d = dst[e];
            float c = dinv[s] * dinv[d];
            const float4 v = *(const float4*)(xw + (size_t)s * 256 + fq);
            float* op = agg + (size_t)d * 256 + fq;
            atomicAdd(op + 0, v.x * c);
            atomicAdd(op + 1, v.y * c);
            atomicAdd(op + 2, v.z * c);
            atomicAdd(op + 3, v.w * c);
        }
    }
}

// ---------- bias + relu in place (over N*64 float4) ----------
__global__ void bias_relu_kernel(float* __restrict__ h, const float* __restrict__ bias, int N) {
    int i = blockIdx.x * blockDim.x + threadIdx.x;
    int total = N * 64;
    if (i < total) {
        int f = (i & 63) << 2;
        float4 v = ((float4*)h)[i];
        const float4 b = *(const float4*)(bias + f);
        v.x = fmaxf(v.x + b.x, 0.0f);
        v.y = fmaxf(v.y + b.y, 0.0f);
        v.z = fmaxf(v.z + b.z, 0.0f);
        v.w = fmaxf(v.w + b.w, 0.0f);
        ((float4*)h)[i] = v;
    }
}

// ---------- gate: gate[n] = h[n,:].gw + gb ; atomicMax per-graph ----------
__global__ __launch_bounds__(256) void gate_kernel(const float* __restrict__ h,
                                                   const float* __restrict__ gw,
                                                   const float* __restrict__ gb,
                                                   const int* __restrict__ batch,
                                                   float* __restrict__ gate,
                                                   unsigned* __restrict__ gmaxu, int N) {
    int lane = threadIdx.x & 31;
    int wave = threadIdx.x >> 5;
    int n = blockIdx.x * 8 + wave;
    if (n >= N) return;
    const float* hp = h + (size_t)n * 256 + lane * 8;
    const float* wp = gw + lane * 8;
    float s = 0.0f;
#pragma unroll
    for (int j = 0; j < 8; ++j) s += hp[j] * wp[j];
#pragma unroll
    for (int off = 16; off > 0; off >>= 1) s += __shfl_xor(s, off, 32);
    if (lane == 0) {
        float g = s + gb[0];
        gate[n] = g;
        atomicMax(&gmaxu[batch[n]], fmap(g));
    }
}

// ---------- exp + denom ----------
__global__ void exp_kernel(const float* __restrict__ gate, const unsigned* __restrict__ gmaxu,
                           const int* __restrict__ batch, float* __restrict__ eexp,
                           float* __restrict__ denom, int N) {
    int n = blockIdx.x * blockDim.x + threadIdx.x;
    if (n < N) {
        int b = batch[n];
        unsigned u = gmaxu[b];
        float gm = (u == 0u) ? 0.0f : funmap(u);   // empty segment -> 0 (matches reference where())
        float e = expf(gate[n] - gm);
        eexp[n] = e;
        atomicAdd(&denom[b], e);
    }
}

// ---------- pooled weighted sum with LDS staging: out[b,:] += alpha[n]*h[n,:] ----------
__global__ __launch_bounds__(256) void pool_kernel(const float* __restrict__ h,
                                                   const float* __restrict__ eexp,
                                                   const float* __restrict__ denom,
                                                   const int* __restrict__ batch,
                                                   float* __restrict__ out, int N) {
    __shared__ float acc[8 * 256];
    const int t = threadIdx.x;                 // feature 0..255 (exclusive per thread)
#pragma unroll
    for (int g = 0; g < 8; ++g) acc[g * 256 + t] = 0.0f;
    int base = blockIdx.x * 64;
    int end = base + 64;
    if (end > N) end = N;
    for (int n = base; n < end; ++n) {
        int b = batch[n];
        float a = eexp[n] / denom[b];
        acc[b * 256 + t] += a * h[(size_t)n * 256 + t];
    }
#pragma unroll
    for (int g = 0; g < 8; ++g) {
        float v = acc[g * 256 + t];
        if (v != 0.0f) atomicAdd(&out[g * 256 + t], v);
    }
}

extern "C" void kernel_launch(void* const* d_in, const int* in_sizes, int n_in,
                              void* d_out, int out_size, void* d_ws, size_t ws_size,
                              hipStream_t stream) {
    const float* x     = (const float*)d_in[0];
    const int*   ei    = (const int*)d_in[1];
    const int*   batch = (const int*)d_in[2];
    const float* W1    = (const float*)d_in[3];
    const float* b1    = (const float*)d_in[4];
    const float* W2    = (const float*)d_in[5];
    const float* b2    = (const float*)d_in[6];
    const float* gw    = (const float*)d_in[7];
    const float* gb    = (const float*)d_in[8];
    float* out = (float*)d_out;

    const int N  = in_sizes[2];
    const int E  = in_sizes[1] / 2;
    const int K1 = in_sizes[0] / N;    // 512
    const int F  = in_sizes[4];        // 256
    const int Bg = out_size / F;       // 8
    const int* src = ei;
    const int* dst = ei + E;

    float* ws      = (float*)d_ws;
    float* dinv    = ws;                       // N (deg accumulator then rsqrt)
    float* gate    = ws + N;                   // N
    float* eexp    = ws + 2 * (size_t)N;       // N
    float* denom   = ws + 3 * (size_t)N;       // Bg
    unsigned* gmx  = (unsigned*)(ws + 3 * (size_t)N + 16);
    float* buf0    = ws + 3 * (size_t)N + 64;  // N*F  (xw1, then xw2)
    float* buf1    = buf0 + (size_t)N * F;     // N*F  (h1, then h2)

    const int elem4 = N * (F / 4);             // float4 count for [N,F]

    // --- init ---
    init_kernel<<<(N + 255) / 256, 256, 0, stream>>>(dinv, denom, gmx, out, N, Bg, out_size);
    deg_kernel<<<(E + 255) / 256, 256, 0, stream>>>(dst, dinv, E);
    rsqrt_kernel<<<(N + 255) / 256, 256, 0, stream>>>(dinv, N);

    // --- layer 1: xw1 = x @ W1 ; agg ; bias+relu ---
    dim3 g1((N + 127) / 128, F / 64);
    gemm_wmma_f32<<<g1, 256, 0, stream>>>(x, W1, buf0, N, K1, F);
    self_init_kernel<<<(elem4 + 255) / 256, 256, 0, stream>>>(buf0, dinv, buf1, N);
    edge_agg_kernel<<<(E + 15) / 16, 256, 0, stream>>>(src, dst, dinv, buf0, buf1, E);
    bias_relu_kernel<<<(elem4 + 255) / 256, 256, 0, stream>>>(buf1, b1, N);

    // --- layer 2: xw2 = h1 @ W2 ; agg ; bias+relu ---
    gemm_wmma_f32<<<g1, 256, 0, stream>>>(buf1, W2, buf0, N, F, F);
    self_init_kernel<<<(elem4 + 255) / 256, 256, 0, stream>>>(buf0, dinv, buf1, N);
    edge_agg_kernel<<<(E + 15) / 16, 256, 0, stream>>>(src, dst, dinv, buf0, buf1, E);
    bias_relu_kernel<<<(elem4 + 255) / 256, 256, 0, stream>>>(buf1, b2, N);

    // --- global attention pooling ---
    gate_kernel<<<(N + 7) / 8, 256, 0, stream>>>(buf1, gw, gb, batch, gate, gmx, N);
    exp_kernel<<<(N + 255) / 256, 256, 0, stream>>>(gate, gmx, batch, eexp, denom, N);
    pool_kernel<<<(N + 63) / 64, 256, 0, stream>>>(buf1, eexp, denom, batch, out, N);
}